// NodeConvolution_5239860101877
// MI455X (gfx1250) — compile-verified
//
#include <hip/hip_runtime.h>

// out[b,s,d] = x[b,2s,d]*w[0,d] + x[b,2s+1,d]*w[1,d]
// B=32, N=4096, D=512, P=2 -> steps=2048, no ragged tail.
// Memory-bound: ~402 MB traffic, ~17us HBM floor @ 23.3 TB/s. WMMA is
// inapplicable (d is not contracted); the win is pure NT streaming.

typedef float f32x4 __attribute__((ext_vector_type(4)));

#define DVEC 128   // D/4 float4 per row
#define WVEC 256   // P*D/4 float4 of weights (4 KB)

__global__ __launch_bounds__(256) void node_pool2_kernel(
    const f32x4* __restrict__ x4,
    const f32x4* __restrict__ w4,
    f32x4* __restrict__ out4,
    unsigned int total_vec)   // out elements / 4 (multiple of 128)
{
    __shared__ f32x4 w_lds[WVEC];

    // ---- Stage the 4 KB weight table into LDS via the CDNA5 async
    // global->LDS DMA path (ASYNCcnt). 256 threads x 16 B = 4 KB.
    {
        const unsigned int t = threadIdx.x;
        unsigned int       lds_addr = (unsigned int)(size_t)&w_lds[t];
        unsigned long long gaddr    = (unsigned long long)(const void*)&w4[t];
        asm volatile("global_load_async_to_lds_b128 %0, %1, off"
                     :: "v"(lds_addr), "v"(gaddr)
                     : "memory");
        asm volatile("s_wait_asynccnt 0" ::: "memory");
    }
    __syncthreads();

    // ---- Each thread owns one float4 column j and strides over rows.
    // grid*block is a multiple of 128, so j is fixed per thread.
    const unsigned int tid     = blockIdx.x * blockDim.x + threadIdx.x;
    const unsigned int j       = tid & (DVEC - 1);
    const unsigned int rstride = (gridDim.x * blockDim.x) >> 7; // rows/step
    const unsigned int rmax    = total_vec >> 7;                // output rows

    // Hoist weights out of LDS into registers once.
    const f32x4 w0 = w_lds[j];
    const f32x4 w1 = w_lds[DVEC + j];

    for (unsigned int r = tid >> 7; r < rmax; r += rstride) {
        const f32x4* px = x4 + (((size_t)r) << 8) + j;   // x row 2r, col j
        // x touched exactly once -> non-temporal (don't pollute L2/WGP$)
        f32x4 a = __builtin_nontemporal_load(px);          // x[b,2s  ,:]
        f32x4 b = __builtin_nontemporal_load(px + DVEC);   // x[b,2s+1,:]

        __builtin_nontemporal_store(a * w0 + b * w1,
                                    out4 + (((size_t)r) << 7) + j);
    }
}

extern "C" void kernel_launch(void* const* d_in, const int* in_sizes, int n_in,
                              void* d_out, int out_size, void* d_ws, size_t ws_size,
                              hipStream_t stream) {
    const f32x4* x4 = (const f32x4*)d_in[0];   // (32, 4096, 512) f32
    const f32x4* w4 = (const f32x4*)d_in[1];   // (2, 512) f32
    f32x4*       o4 = (f32x4*)d_out;           // (32, 2048, 512) f32

    const unsigned int total_vec = (unsigned int)(out_size / 4); // 8,388,608
    const unsigned int block     = 256;
    unsigned int grid            = 8192;       // grid-stride, 4 rows/thread
    const unsigned int max_grid  = (total_vec + block - 1) / block;
    if (grid > max_grid) grid = max_grid;

    node_pool2_kernel<<<grid, block, 0, stream>>>(x4, w4, o4, total_vec);
}